// MySoftBCELoss_19335942766644
// MI455X (gfx1250) — compile-verified
//
#include <hip/hip_runtime.h>
#include <math.h>

typedef __attribute__((ext_vector_type(2))) float v2f;
typedef __attribute__((ext_vector_type(8))) float v8f;

#define B_ROWS    (1 << 20)
#define C_CLS     32
#define EPS_F     1e-7f
#define NBLOCKS   2048
#define NTHREADS  256
#define WAVES_PER_BLOCK (NTHREADS / 32)
#define TOTAL_WAVES     (NBLOCKS * WAVES_PER_BLOCK)   // 16384
#define NTILES          (B_ROWS / 16)                 // 65536
#define TILES_PER_WAVE  (NTILES / TOTAL_WAVES)        // 4

#define LOG2E_F 1.4426950408889634f
#define LN2_F   0.6931471805599453f

// lane <-> lane^16 swap, pure VALU (v_permlanex16_b32 with identity selects)
__device__ __forceinline__ int xor16i(int v) {
    return __builtin_amdgcn_permlanex16(v, v, 0x76543210, 0xFEDCBA98, false, false);
}
__device__ __forceinline__ float xor16f(float v) {
    int r = xor16i(__builtin_bit_cast(int, v));
    return __builtin_bit_cast(float, r);
}

__global__ __launch_bounds__(NTHREADS)
void softbce_partial_kernel(const float* __restrict__ logits,
                            const float* __restrict__ target,
                            float* __restrict__ partials)
{
    const int lane   = threadIdx.x & 31;
    const int waveId = blockIdx.x * WAVES_PER_BLOCK + (threadIdx.x >> 5);
    const int half   = lane >> 4;    // 0 -> cols 0..15, 1 -> cols 16..31
    const int row16  = lane & 15;    // row within the 16-row tile (== WMMA M)

    v8f acc = {};                    // WMMA f32 D accumulator (row sums, log2 units)
    v2f onesB = {1.0f, 1.0f};        // B = all ones -> D row = sum over K of A row
    float s1_acc = 0.0f;             // label!=0 branch accumulator (log2 units)

    for (int t = 0; t < TILES_PER_WAVE; ++t) {
        const int tile = t * TOTAL_WAVES + waveId;
        const int base = tile * (16 * C_CLS) + row16 * C_CLS + half * 16;

        // Prefetch next tile (speculative; silently dropped if OOB)
        __builtin_prefetch(logits + base + 16 * C_CLS * TOTAL_WAVES, 0, 1);
        __builtin_prefetch(target + base + 16 * C_CLS * TOTAL_WAVES, 0, 1);

        // 16 contiguous floats per lane from each array: 4 x b128 loads each
        float lg[16], tg[16];
        const float4* lp = reinterpret_cast<const float4*>(logits + base);
        const float4* tp = reinterpret_cast<const float4*>(target + base);
#pragma unroll
        for (int q = 0; q < 4; ++q) {
            float4 a = lp[q];
            float4 b = tp[q];
            lg[4*q+0] = a.x; lg[4*q+1] = a.y; lg[4*q+2] = a.z; lg[4*q+3] = a.w;
            tg[4*q+0] = b.x; tg[4*q+1] = b.y; tg[4*q+2] = b.z; tg[4*q+3] = b.w;
        }

        // Elementwise soft-BCE terms (log2 units) + in-lane argmax scan.
        // pred in [eps, 1-eps] -> raw v_exp_f32/v_log_f32/v_rcp_f32 are safe
        // (no inf/denorm fixups needed).
        float e[16];
        float log1mp0 = 0.0f;                 // log2(1-pred) at column 0 (lower lane)
        float maxT = -INFINITY, lpAt = 0.0f;  // lpAt in log2 units
        int   idx  = 0;
#pragma unroll
        for (int j = 0; j < 16; ++j) {
            float x   = lg[j];
            float em  = __builtin_amdgcn_exp2f(-x * LOG2E_F);     // e^{-x}
            float p   = __builtin_amdgcn_rcpf(1.0f + em);         // sigmoid
            p         = fminf(fmaxf(p, EPS_F), 1.0f - EPS_F);
            float lp2  = __builtin_amdgcn_logf(p);                // log2(p)
            float l1p2 = __builtin_amdgcn_logf(1.0f - p);         // log2(1-p)
            float tv   = tg[j];
            e[j] = fmaf(tv, lp2 - l1p2, l1p2);   // t*log2 p + (1-t)*log2(1-p)
            log1mp0 = (half == 0 && j == 0) ? l1p2 : log1mp0;     // compile-time j
            bool gt = tv > maxT;                 // strict > keeps first occurrence
            idx  = gt ? (half * 16 + j) : idx;
            lpAt = gt ? lp2 : lpAt;
            maxT = gt ? tv  : maxT;
        }

        // Combine the two half-lanes of each row (pure VALU, EXEC stays full)
        float oMax = xor16f(maxT);
        int   oIdx = xor16i(idx);
        float oLp  = xor16f(lpAt);
        float loM = (half == 0) ? maxT : oMax;
        int   loI = (half == 0) ? idx  : oIdx;
        float loL = (half == 0) ? lpAt : oLp;
        float hiM = (half == 0) ? oMax : maxT;
        int   hiI = (half == 0) ? oIdx : idx;
        float hiL = (half == 0) ? oLp  : lpAt;
        bool  hiWins = hiM > loM;                // ties -> lower column index
        float rM = hiWins ? hiM : loM;
        int   rI = hiWins ? hiI : loI;
        float rL = hiWins ? hiL : loL;

        // label != 0 branch (NEG_WEIGHT == 1.0); only lower lane owns col 0
        s1_acc += ((half == 0) && (rI != 0)) ? fmaf(rM, rL, log1mp0) : 0.0f;

        // label == 0 branch: row mean via WMMA against ones; zero other rows
        float scale = (rI == 0) ? (1.0f / 32.0f) : 0.0f;
#pragma unroll
        for (int k = 0; k < 8; ++k) {
            v2f a = { e[2*k] * scale, e[2*k + 1] * scale };
            acc = __builtin_amdgcn_wmma_f32_16x16x4_f32(
                false, a, false, onesB, (short)0, acc, false, false);
        }
    }

    // Each row-sum is replicated across 16 lanes (columns) in D -> divide by 16,
    // and convert log2 -> ln once.
    float psum = 0.0f;
#pragma unroll
    for (int v = 0; v < 8; ++v) psum += acc[v];
    float p = (psum * (1.0f / 16.0f) + s1_acc) * LN2_F;

    // Wave32 reduction (once per kernel; DS path is fine here)
#pragma unroll
    for (int off = 16; off > 0; off >>= 1) p += __shfl_xor(p, off, 32);

    __shared__ float smem[WAVES_PER_BLOCK];
    if (lane == 0) smem[threadIdx.x >> 5] = p;
    __syncthreads();
    if (threadIdx.x == 0) {
        float s = 0.0f;
#pragma unroll
        for (int w = 0; w < WAVES_PER_BLOCK; ++w) s += smem[w];
        partials[blockIdx.x] = s;
    }
}

__global__ __launch_bounds__(NTHREADS)
void softbce_finalize_kernel(const float* __restrict__ partials,
                             float* __restrict__ out)
{
    __shared__ float sm[NTHREADS];
    float s = 0.0f;
    for (int i = threadIdx.x; i < NBLOCKS; i += NTHREADS) s += partials[i];
    sm[threadIdx.x] = s;
    __syncthreads();
    for (int stride = NTHREADS / 2; stride > 0; stride >>= 1) {
        if (threadIdx.x < stride) sm[threadIdx.x] += sm[threadIdx.x + stride];
        __syncthreads();
    }
    if (threadIdx.x == 0) out[0] = -sm[0] / (float)B_ROWS;
}

extern "C" void kernel_launch(void* const* d_in, const int* in_sizes, int n_in,
                              void* d_out, int out_size, void* d_ws, size_t ws_size,
                              hipStream_t stream)
{
    (void)in_sizes; (void)n_in; (void)out_size; (void)ws_size;
    const float* logits = (const float*)d_in[0];
    const float* target = (const float*)d_in[1];
    float* partials = (float*)d_ws;          // NBLOCKS floats of scratch
    float* out      = (float*)d_out;

    softbce_partial_kernel<<<NBLOCKS, NTHREADS, 0, stream>>>(logits, target, partials);
    softbce_finalize_kernel<<<1, NTHREADS, 0, stream>>>(partials, out);
}